// ALiBi_8538394985269
// MI455X (gfx1250) — compile-verified
//
#include <hip/hip_runtime.h>

// ALiBi bias subtraction: out = scores - slopes[h] * ((pos[i]+off) - (pos[j]+off))
// Shapes (from reference): scores [B=2, H=16, S=2048, S=2048] fp32.
// Pure streaming elementwise op -> memory-bound (AI = 0.25 FLOP/byte).
// Strategy: 128-bit vector loads/stores with non-temporal cache policy
// (CDNA5 TH=NT) so the 1 GB one-shot stream does not churn the 192 MB L2;
// 8-row unroll per thread for MLP; column positions hoisted out of the row loop.

typedef __attribute__((ext_vector_type(4))) float v4f;

namespace {
constexpr int S    = 2048;  // sequence length (rows == cols)
constexpr int H    = 16;    // heads
constexpr int B    = 2;     // batch
constexpr int ROWS = 8;     // rows per thread (unroll depth / MLP)
constexpr int TPB  = 256;   // threads per block = 8 waves (wave32)
}

__global__ __launch_bounds__(TPB)
void alibi_sub_kernel(const float* __restrict__ scores,
                      const float* __restrict__ slopes,
                      const float* __restrict__ positions,
                      const int*   __restrict__ offset_p,
                      float* __restrict__ out)
{
    // Column slice owned by this thread: 4 consecutive floats (one b128).
    const int j  = (blockIdx.x * TPB + (int)threadIdx.x) * 4;
    const int i0 = blockIdx.y * ROWS;          // first row of this block's tile
    const int bh = blockIdx.z;                 // fused (batch, head)
    const int h  = bh & (H - 1);

    const float off   = (float)(*offset_p);    // scalar, cache-resident
    const float slope = slopes[h];             // block-uniform -> scalar path

    // Column positions: loaded once, reused across all ROWS rows.
    v4f pj = *(const v4f*)(positions + j);
    pj.x += off; pj.y += off; pj.z += off; pj.w += off;

    const size_t base = ((size_t)bh * S + (size_t)i0) * (size_t)S + (size_t)j;

    // Issue all row loads first (independent b128s -> deep MLP), NT policy.
    v4f sc[ROWS];
#pragma unroll
    for (int r = 0; r < ROWS; ++r) {
        sc[r] = __builtin_nontemporal_load(
            (const v4f*)(scores + base + (size_t)r * S));
    }

#pragma unroll
    for (int r = 0; r < ROWS; ++r) {
        const float pi = positions[i0 + r] + off;  // row-uniform
        v4f o;
        o.x = sc[r].x - slope * (pi - pj.x);
        o.y = sc[r].y - slope * (pi - pj.y);
        o.z = sc[r].z - slope * (pi - pj.z);
        o.w = sc[r].w - slope * (pi - pj.w);
        __builtin_nontemporal_store(o, (v4f*)(out + base + (size_t)r * S));
    }
}

extern "C" void kernel_launch(void* const* d_in, const int* in_sizes, int n_in,
                              void* d_out, int out_size, void* d_ws, size_t ws_size,
                              hipStream_t stream) {
    (void)in_sizes; (void)n_in; (void)out_size; (void)d_ws; (void)ws_size;

    const float* scores    = (const float*)d_in[0];  // [B,H,S,S] fp32
    const float* slopes    = (const float*)d_in[1];  // [H] fp32
    const float* positions = (const float*)d_in[2];  // [CONTEXT_LENGTH] fp32
    const int*   offset_p  = (const int*)d_in[3];    // scalar int
    float*       out       = (float*)d_out;          // [B,H,S,S] fp32

    dim3 block(TPB, 1, 1);
    dim3 grid(S / (TPB * 4),   // 2  column tiles of 1024 floats
              S / ROWS,        // 256 row tiles
              B * H);          // 32 (batch, head) slices

    alibi_sub_kernel<<<grid, block, 0, stream>>>(scores, slopes, positions,
                                                 offset_p, out);
}